// BandSplit_10179072492132
// MI455X (gfx1250) — compile-verified
//
#include <hip/hip_runtime.h>

// ---------------------------------------------------------------------------
// BSRNN band-split for MI455X (gfx1250, wave32).
//   y[b,:,band,t] = rstd * (W'_band @ h_band[b,:,t]) + (bias + u - mu*rstd*v)
// W' = W*gamma folded once; GroupNorm folded into the GEMM epilogue so the
// WMMA GEMM runs on raw spectrogram data. f32 WMMA (16x16x4) keeps full
// precision; problem is ~balanced compute/memory, so the key optimization is
// software-pipelined global->reg->LDS staging to keep HBM requests in flight
// under the WMMA stream.
// ---------------------------------------------------------------------------

#define TOTAL_CH 8196
#define NBANDS   31
#define OUTC     128
#define TDIM     512
#define FDIM     2049
#define BDIM     8

typedef __attribute__((ext_vector_type(2))) float v2f;
typedef __attribute__((ext_vector_type(4))) float v4f;
typedef __attribute__((ext_vector_type(8))) float v8f;

__device__ __constant__ int d_start[NBANDS] = {
    0, 25, 50, 75, 100, 125, 150, 175, 200, 225,
    250, 300, 350, 400, 450, 500, 550, 600, 650, 700, 750, 800,
    850, 950, 1050, 1150, 1250, 1350, 1450, 1550,
    1650};
__device__ __constant__ int d_width[NBANDS] = {
    25, 25, 25, 25, 25, 25, 25, 25, 25, 25,
    50, 50, 50, 50, 50, 50, 50, 50, 50, 50, 50, 50,
    100, 100, 100, 100, 100, 100, 100, 100,
    399};

// ---------------- pass 0: W' = W * gamma (columnwise) ----------------------
__global__ __launch_bounds__(256) void bs_prep_wp(const float* __restrict__ W,
                                                  const float* __restrict__ gamma,
                                                  float* __restrict__ Wp, int n) {
  int i = blockIdx.x * 256 + threadIdx.x;
  if (i < n) {
    int c = i % TOTAL_CH;
    Wp[i] = W[i] * gamma[c];
  }
}

// ---------------- pass 0b: u = W.beta, v = W.gamma per band ----------------
__global__ __launch_bounds__(128) void bs_prep_uv(const float* __restrict__ W,
                                                  const float* __restrict__ gamma,
                                                  const float* __restrict__ beta,
                                                  float* __restrict__ u,
                                                  float* __restrict__ v) {
  const int band = blockIdx.x;
  const int o = threadIdx.x;
  const int s = d_start[band], w = d_width[band];
  const int off = 4 * s, ch = 4 * w;
  const float* wr = W + (size_t)o * TOTAL_CH + off;
  float uu = 0.f, vv = 0.f;
  for (int c = 0; c < ch; ++c) {
    float wv = wr[c];
    uu += wv * beta[off + c];
    vv += wv * gamma[off + c];
  }
  u[band * OUTC + o] = uu;
  v[band * OUTC + o] = vv;
}

// ---------------- pass 1a: partial sums / sumsq per (b,band,chunk) ---------
// Loads float2 {ri=0, ri=1} per (cc,f',t): fully coalesced global_load_b64,
// both components feed the same statistics.
__global__ __launch_bounds__(256) void bs_stats_partial(const float* __restrict__ x,
                                                        float* __restrict__ partials) {
  const int xg = blockIdx.x;      // b*31 + band
  const int chunk = blockIdx.y;   // 0..15
  const int bb = xg / NBANDS, band = xg % NBANDS;
  const int s = d_start[band], w = d_width[band];
  const int len2 = 2 * w * TDIM;  // (cc,f') pairs x frames
  float sum = 0.f, sum2 = 0.f;
  for (int i = chunk * 256 + threadIdx.x; i < len2; i += 16 * 256) {
    int p = i >> 9;   // cc*w + f'
    int t = i & 511;  // frame
    int cc = p / w;
    int fp = p - cc * w;
    v2f xv = *(const v2f*)(x + (size_t)(((bb * 2 + cc) * FDIM + s + fp) * TDIM + t) * 2);
    sum += xv[0] + xv[1];
    sum2 += xv[0] * xv[0] + xv[1] * xv[1];
  }
  __shared__ float sd[512];
  sd[threadIdx.x] = sum;
  sd[256 + threadIdx.x] = sum2;
  __syncthreads();
  for (int st = 128; st > 0; st >>= 1) {
    if ((int)threadIdx.x < st) {
      sd[threadIdx.x] += sd[threadIdx.x + st];
      sd[256 + threadIdx.x] += sd[256 + threadIdx.x + st];
    }
    __syncthreads();
  }
  if (threadIdx.x == 0) {
    partials[(xg * 16 + chunk) * 2 + 0] = sd[0];
    partials[(xg * 16 + chunk) * 2 + 1] = sd[256];
  }
}

// ---------------- pass 1b: rstd + fused epilogue bias fb -------------------
__global__ __launch_bounds__(128) void bs_stats_final(const float* __restrict__ partials,
                                                      const float* __restrict__ u,
                                                      const float* __restrict__ v,
                                                      const float* __restrict__ bias,
                                                      float* __restrict__ fb,
                                                      float* __restrict__ stats) {
  const int xg = blockIdx.x;  // b*31 + band
  const int band = xg % NBANDS;
  const int w = d_width[band];
  float sum = 0.f, sum2 = 0.f;
  for (int j = 0; j < 16; ++j) {
    sum += partials[(xg * 16 + j) * 2 + 0];
    sum2 += partials[(xg * 16 + j) * 2 + 1];
  }
  const float n = (float)(4 * w * TDIM);
  const float mu = sum / n;
  float var = sum2 / n - mu * mu;
  const float rstd = rsqrtf(var + 1e-5f);
  const int o = threadIdx.x;
  fb[xg * OUTC + o] = bias[band * OUTC + o] + u[band * OUTC + o]
                    - mu * rstd * v[band * OUTC + o];
  if (o == 0) stats[xg] = rstd;
}

// ---------------- pass 2: fused GEMM via f32 WMMA --------------------------
// block = 256 threads (8 waves). Tile: M=128 (8 waves x 16), N=64 frames,
// K tiled by 64 with zero padding. Software pipeline: global loads of chunk
// k+1 are issued into registers right after LDS for chunk k becomes valid,
// so HBM latency hides under the 16 K-step WMMA sequence. LDS layouts
// [row][k], pad 68 floats -> fragment ds_load_b64 is bank-conflict free.
__global__ __launch_bounds__(256) void bs_band_gemm(const float* __restrict__ x,
                                                    const float* __restrict__ Wp,
                                                    const float* __restrict__ fb,
                                                    const float* __restrict__ stats,
                                                    float* __restrict__ out) {
  const int band = blockIdx.z;
  const int bb = blockIdx.y;
  const int t0 = blockIdx.x * 64;
  const int s = d_start[band], w = d_width[band];
  const int off = 4 * s, ch = 4 * w;

  __shared__ float ldsA[128 * 68];  // W' tile [o][k]
  __shared__ float ldsB[64 * 68];   // h  tile [t][k]

  const int tid = threadIdx.x;
  const int lane = tid & 31;
  const int wid = tid >> 5;
  const int m0 = wid * 16;
  const int lm = lane & 15;
  const int lh = lane >> 4;

  v8f acc[4];
#pragma unroll
  for (int j = 0; j < 4; ++j)
    acc[j] = (v8f){0.f, 0.f, 0.f, 0.f, 0.f, 0.f, 0.f, 0.f};

  const int aoff = (m0 + lm) * 68 + 2 * lh;
  int boff[4];
#pragma unroll
  for (int j = 0; j < 4; ++j) boff[j] = (j * 16 + lm) * 68 + 2 * lh;

  v4f aReg[8];     // A stage: 128x64 floats / 256 threads, float4 slots
  float bReg[16];  // B stage: 64x64 floats / 256 threads

  auto prefetchA = [&](int k0) {
#pragma unroll
    for (int j = 0; j < 8; ++j) {
      const int i = tid + j * 256;   // 0..2047
      const int o = i >> 4;
      const int q = (i & 15) << 2;
      v4f val = (v4f){0.f, 0.f, 0.f, 0.f};
      if (k0 + q < ch)
        val = *(const v4f*)(Wp + (size_t)o * TOTAL_CH + off + k0 + q);
      aReg[j] = val;
    }
  };
  auto prefetchB = [&](int k0) {
#pragma unroll
    for (int j = 0; j < 16; ++j) {
      const int i = tid + j * 256;   // 0..4095
      const int kq = i >> 6;
      const int tt = i & 63;
      const int c = k0 + kq;
      float vv = 0.f;
      if (c < ch) {
        int ri = c / (2 * w);
        int rem = c - ri * 2 * w;
        int cc = rem / w;
        int fp = rem - cc * w;
        vv = x[(size_t)(((bb * 2 + cc) * FDIM + s + fp) * TDIM + (t0 + tt)) * 2 + ri];
      }
      bReg[j] = vv;
    }
  };

  // prologue: chunk 0 in flight
  prefetchA(0);
  prefetchB(0);

  for (int k0 = 0; k0 < ch; k0 += 64) {
    __syncthreads();  // previous-iteration LDS readers done (no-op first iter)
    // drain prefetch registers into LDS
#pragma unroll
    for (int j = 0; j < 8; ++j) {
      const int i = tid + j * 256;
      *(v4f*)(ldsA + (i >> 4) * 68 + ((i & 15) << 2)) = aReg[j];
    }
#pragma unroll
    for (int j = 0; j < 16; ++j) {
      const int i = tid + j * 256;
      ldsB[(i & 63) * 68 + (i >> 6)] = bReg[j];
    }
    __syncthreads();

    // issue next chunk's global loads; s_wait lands before next LDS drain
    const int k1 = k0 + 64;
    if (k1 < ch) {
      prefetchA(k1);
      prefetchB(k1);
    }

    // 16 K-steps of f32 WMMA from LDS
#pragma unroll
    for (int kk = 0; kk < 64; kk += 4) {
      v2f a = *(const v2f*)(ldsA + aoff + kk);
#pragma unroll
      for (int j = 0; j < 4; ++j) {
        v2f bf = *(const v2f*)(ldsB + boff[j] + kk);
        acc[j] = __builtin_amdgcn_wmma_f32_16x16x4_f32(
            false, a, false, bf, (short)0, acc[j], false, false);
      }
    }
  }

  // epilogue: y = rstd*acc + fb ; C/D layout: VGPR r -> M=r+8*lh, lane%16 -> N
  // non-temporal stores: output is write-once, keep x/W' resident in L2
  const int sb = bb * NBANDS + band;
  const float rstd = stats[sb];
  const float* fbp = fb + sb * OUTC + m0 + 8 * lh;
  v4f f0 = *(const v4f*)(fbp);
  v4f f1 = *(const v4f*)(fbp + 4);
#pragma unroll
  for (int r = 0; r < 8; ++r) {
    const int o = m0 + 8 * lh + r;
    const float fbr = (r < 4) ? f0[r] : f1[r - 4];
#pragma unroll
    for (int j = 0; j < 4; ++j) {
      __builtin_nontemporal_store(
          rstd * acc[j][r] + fbr,
          out + ((bb * OUTC + o) * NBANDS + band) * TDIM + t0 + j * 16 + lm);
    }
  }
}

// ---------------------------------------------------------------------------
extern "C" void kernel_launch(void* const* d_in, const int* in_sizes, int n_in,
                              void* d_out, int out_size, void* d_ws, size_t ws_size,
                              hipStream_t stream) {
  const float* x     = (const float*)d_in[0];  // [8,2,2049,512,2]
  const float* gamma = (const float*)d_in[1];  // [8196]
  const float* beta  = (const float*)d_in[2];  // [8196]
  const float* W     = (const float*)d_in[3];  // [128,8196]
  const float* bias  = (const float*)d_in[4];  // [31,128]
  float* out = (float*)d_out;                  // [8,128,31,512]

  float* ws = (float*)d_ws;
  float* Wp       = ws;                     // 128*8196 = 1,049,088
  float* u        = Wp + 128 * TOTAL_CH;    // 31*128 = 3968
  float* v        = u + NBANDS * OUTC;      // 3968
  float* partials = v + NBANDS * OUTC;      // 248*16*2 = 7936
  float* stats    = partials + 248 * 32;    // 248
  float* fb       = stats + 248;            // 8*31*128 = 31744
  // total ~4.4 MB of workspace

  bs_prep_wp<<<(128 * TOTAL_CH + 255) / 256, 256, 0, stream>>>(W, gamma, Wp,
                                                               128 * TOTAL_CH);
  bs_prep_uv<<<NBANDS, 128, 0, stream>>>(W, gamma, beta, u, v);
  bs_stats_partial<<<dim3(BDIM * NBANDS, 16), 256, 0, stream>>>(x, partials);
  bs_stats_final<<<BDIM * NBANDS, 128, 0, stream>>>(partials, u, v, bias, fb, stats);
  bs_band_gemm<<<dim3(TDIM / 64, BDIM, NBANDS), 256, 0, stream>>>(x, Wp, fb, stats, out);
}